// PANUMPooling_8787503088315
// MI455X (gfx1250) — compile-verified
//
#include <hip/hip_runtime.h>
#include <cstdint>

// Problem constants (from reference)
#define NTOT 16384      // B * NPG
#define BGR  8          // graphs
#define NPG  2048       // nodes per graph
#define KTOP 1024       // kept per graph
#define FDIM 128        // feature dim
#define NEDGE 262144    // edges
#define NKEEP (BGR * KTOP)   // 8192

// d_out layout (outputs concatenated flat, all as float32)
#define OFF_X      0
#define OFF_EI     (NKEEP * FDIM)            // 1048576
#define OFF_EW     (OFF_EI + 2 * NEDGE)      // 1572864
#define OFF_BATCH  (OFF_EW + NEDGE)          // 1835008
#define OFF_PERM   (OFF_BATCH + NKEEP)       // 1843200
#define OFF_SCORE  (OFF_PERM + NKEEP)        // 1851392

typedef __attribute__((ext_vector_type(4))) unsigned int v4u_t;
typedef __attribute__((ext_vector_type(8))) int v8i_t;
typedef __attribute__((ext_vector_type(4))) int v4i_t;

// ---------------------------------------------------------------------------
// Kernel 1: per-graph score = tanh(diag(UM)), bitonic top-K (exact stable
// descending argsort semantics), write perm/score/batch, build node_map.
// One workgroup (1024 threads, 32 waves) per graph.
// Diagonal is fetched with the Tensor Data Mover: 2D tile, 1 elem/line,
// line stride = N+1 floats -> single async DMA into LDS (TENSORcnt).
// ---------------------------------------------------------------------------
__global__ __launch_bounds__(1024)
void panum_topk_kernel(const float* __restrict__ UM,
                       float* __restrict__ out,
                       int* __restrict__ node_map,
                       int* __restrict__ perm_ws) {
  __shared__ float s_key[NPG];
  __shared__ int   s_idx[NPG];
  const int b   = blockIdx.x;
  const int tid = threadIdx.x;

#if defined(__HIP_DEVICE_COMPILE__) && defined(__gfx1250__) && \
    __has_builtin(__builtin_amdgcn_tensor_load_to_lds) && \
    __has_builtin(__builtin_amdgcn_s_wait_tensorcnt)
  if ((tid >> 5) == 0) {   // wave 0 of the workgroup issues the DMA once
    const uint64_t gaddr =
        (uint64_t)(uintptr_t)UM + (uint64_t)b * NPG * (uint64_t)(NTOT + 1) * 4ull;
    const unsigned lds_off = (unsigned)(uintptr_t)(void*)&s_key[0];

    // D# group 0: count=1 | lds_addr | global_addr[56:0] | type=2
    v4u_t g0;
    g0[0] = 1u;
    g0[1] = lds_off;
    g0[2] = (unsigned)(gaddr & 0xFFFFFFFFu);
    g0[3] = (unsigned)((gaddr >> 32) & 0x1FFFFFFu) | (2u << 30);

    // D# group 1: data_size=4B; tensor_dim0=N+1, tensor_dim1=NPG;
    // tile_dim0=1, tile_dim1=NPG; tensor_dim0_stride=N+1 (diagonal walk)
    const unsigned td0 = NTOT + 1, td1 = NPG, t0 = 1, t1 = NPG;
    const uint64_t st0 = (uint64_t)(NTOT + 1);
    v8i_t g1;
    g1[0] = (int)(2u << 16);
    g1[1] = (int)((td0 & 0xFFFFu) << 16);
    g1[2] = (int)(((td0 >> 16) & 0xFFFFu) | ((td1 & 0xFFFFu) << 16));
    g1[3] = (int)(((td1 >> 16) & 0xFFFFu) | ((t0 & 0xFFFFu) << 16));
    g1[4] = (int)(t1 & 0xFFFFu);
    g1[5] = (int)(st0 & 0xFFFFFFFFull);
    g1[6] = (int)((st0 >> 32) & 0xFFFFull);
    g1[7] = 0;

    v4i_t gz; gz[0] = 0; gz[1] = 0; gz[2] = 0; gz[3] = 0;   // 2D: dims 2..4 unused
#if __clang_major__ >= 23
    v8i_t g4; g4[0]=0; g4[1]=0; g4[2]=0; g4[3]=0; g4[4]=0; g4[5]=0; g4[6]=0; g4[7]=0;
    __builtin_amdgcn_tensor_load_to_lds(g0, g1, gz, gz, g4, 0);
#else
    __builtin_amdgcn_tensor_load_to_lds(g0, g1, gz, gz, 0);
#endif
    __builtin_amdgcn_s_wait_tensorcnt(0);
  }
  __syncthreads();
  for (int i = tid; i < NPG; i += 1024) {
    s_key[i] = tanhf(s_key[i]);   // TDM left raw diag values in LDS
    s_idx[i] = i;
  }
#else
  for (int i = tid; i < NPG; i += 1024) {
    const int n = b * NPG + i;
    s_key[i] = tanhf(UM[(size_t)n * NTOT + n]);
    s_idx[i] = i;
  }
#endif
  __syncthreads();

  // Bitonic sort: descending score, ties -> ascending index (== stable argsort)
  for (int k = 2; k <= NPG; k <<= 1) {
    for (int j = k >> 1; j > 0; j >>= 1) {
      for (int i = tid; i < NPG; i += 1024) {
        const int p = i ^ j;
        if (p > i) {
          const float ka = s_key[i], kb = s_key[p];
          const int   ia = s_idx[i], ib = s_idx[p];
          const bool a_first = (ka > kb) || (ka == kb && ia < ib);
          const bool want_a_first = ((i & k) == 0);
          if (a_first != want_a_first) {
            s_key[i] = kb; s_key[p] = ka;
            s_idx[i] = ib; s_idx[p] = ia;
          }
        }
      }
      __syncthreads();
    }
  }

  // node_map: -1 for dropped, new id for kept (this block owns its graph's slice)
  for (int i = tid; i < NPG; i += 1024) node_map[b * NPG + i] = -1;
  __syncthreads();

  {
    const int j   = tid;                 // exactly K=1024 threads -> K kept nodes
    const int g   = b * NPG + s_idx[j];  // global node id
    const int nid = b * KTOP + j;        // new (pooled) node id
    node_map[g]  = nid;
    perm_ws[nid] = g;
    out[OFF_PERM  + nid] = (float)g;
    out[OFF_SCORE + nid] = s_key[j];
    out[OFF_BATCH + nid] = (float)b;
  }
}

// ---------------------------------------------------------------------------
// Kernel 2: x_new[row, :] = x[perm[row], :] * score_perm[row]
// One wave32 per row: 32 lanes x float4 = 128 floats, fully coalesced.
// ---------------------------------------------------------------------------
__global__ __launch_bounds__(256)
void panum_gather_kernel(const float* __restrict__ x,
                         const int* __restrict__ perm_ws,
                         const float* __restrict__ score_perm,
                         float* __restrict__ out_x) {
  const int t    = blockIdx.x * 256 + threadIdx.x;
  const int row  = t >> 5;       // wave-uniform
  const int lane = t & 31;
  const int   p = perm_ws[row];
  const float s = score_perm[row];
  float4 v = ((const float4*)(x + (size_t)p * FDIM))[lane];
  v.x *= s; v.y *= s; v.z *= s; v.w *= s;
  ((float4*)(out_x + (size_t)row * FDIM))[lane] = v;
}

// ---------------------------------------------------------------------------
// Kernel 3: edge remap through node_map; invalid edges -> -1 / weight 0.
// ---------------------------------------------------------------------------
__global__ __launch_bounds__(256)
void panum_edge_kernel(const int* __restrict__ ei,
                       const float* __restrict__ ew,
                       const int* __restrict__ node_map,
                       float* __restrict__ out_ei,
                       float* __restrict__ out_ew) {
  const int e = blockIdx.x * 256 + threadIdx.x;
  if (e >= NEDGE) return;
  const int r = node_map[ei[e]];
  const int c = node_map[ei[NEDGE + e]];
  const bool m = (r >= 0) && (c >= 0);
  out_ei[e]         = (float)(m ? r : -1);
  out_ei[NEDGE + e] = (float)(m ? c : -1);
  out_ew[e]         = m ? ew[e] : 0.0f;
}

extern "C" void kernel_launch(void* const* d_in, const int* in_sizes, int n_in,
                              void* d_out, int out_size, void* d_ws, size_t ws_size,
                              hipStream_t stream) {
  (void)in_sizes; (void)n_in; (void)out_size; (void)ws_size;
  const float* x  = (const float*)d_in[0];
  const float* UM = (const float*)d_in[1];
  const int*   ei = (const int*)d_in[2];
  const float* ew = (const float*)d_in[3];
  // d_in[4] (batch) is derivable: batch_new[nid] = nid / KTOP

  float* out      = (float*)d_out;
  int*   node_map = (int*)d_ws;           // NTOT ints  (64 KB)
  int*   perm_ws  = node_map + NTOT;      // NKEEP ints (32 KB)

  panum_topk_kernel<<<BGR, 1024, 0, stream>>>(UM, out, node_map, perm_ws);
  panum_gather_kernel<<<(NKEEP * 32) / 256, 256, 0, stream>>>(
      x, perm_ws, out + OFF_SCORE, out + OFF_X);
  panum_edge_kernel<<<(NEDGE + 255) / 256, 256, 0, stream>>>(
      ei, ew, node_map, out + OFF_EI, out + OFF_EW);
}